// TGCNWithInteractions_30666066493879
// MI455X (gfx1250) — compile-verified
//
#include <hip/hip_runtime.h>
#include <math.h>

#define N_NODES 2048
#define E_EDGES 65536
#define SEQ_LEN 5
#define IN_DIM  32
#define HID     64
#define OUT_DIM 16

typedef float v2f __attribute__((ext_vector_type(2)));
typedef float v8f __attribute__((ext_vector_type(8)));

// ---------------------------------------------------------------------------
// small utility kernels
// ---------------------------------------------------------------------------
__global__ void k_zero(float* p, int n) {
    int i = blockIdx.x * blockDim.x + threadIdx.x;
    if (i < n) p[i] = 0.0f;
}

__global__ void k_deg(const int* __restrict__ dst, const float* __restrict__ ew,
                      float* __restrict__ deg) {
    int e = blockIdx.x * blockDim.x + threadIdx.x;
    if (e < E_EDGES) atomicAdd(&deg[dst[e]], ew[e]);
}

__global__ void k_dinv(const float* __restrict__ deg, float* __restrict__ dinv,
                       float* __restrict__ dself) {
    int n = blockIdx.x * blockDim.x + threadIdx.x;
    if (n < N_NODES) {
        float d = rsqrtf(deg[n] + 1.0f);
        dinv[n]  = d;
        dself[n] = d * d;
    }
}

__global__ void k_norm(const int* __restrict__ src, const int* __restrict__ dst,
                       const float* __restrict__ ew, const float* __restrict__ dinv,
                       float* __restrict__ norm) {
    int e = blockIdx.x * blockDim.x + threadIdx.x;
    if (e < E_EDGES) norm[e] = dinv[src[e]] * ew[e] * dinv[dst[e]];
}

// g[n,c] = dself[n]*hw[n,c] + bias[c]   (self-loop term of GCN + bias)
__global__ void k_init_g(const float* __restrict__ hw, const float* __restrict__ dself,
                         const float* __restrict__ bias, float* __restrict__ g) {
    int i = blockIdx.x * blockDim.x + threadIdx.x;
    if (i < N_NODES * HID) {
        int n = i >> 6, c = i & 63;
        g[i] = dself[n] * hw[i] + bias[c];
    }
}

// g[dst,c] += norm[e]*hw[src,c]  (edge scatter; g is L2-resident, 512KB)
__global__ void k_edge_agg(const int* __restrict__ src, const int* __restrict__ dst,
                           const float* __restrict__ norm, const float* __restrict__ hw,
                           float* __restrict__ g) {
    int i = blockIdx.x * blockDim.x + threadIdx.x;
    if (i < E_EDGES * HID) {
        int e = i >> 6, c = i & 63;
        atomicAdd(&g[dst[e] * HID + c], norm[e] * hw[src[e] * HID + c]);
    }
}

// hs[n, t*HID + c] = (1 - Z[n,c]) * Ht[n,c]
__global__ void k_combine(const float* __restrict__ Z, const float* __restrict__ Ht,
                          float* __restrict__ hs, int t) {
    int i = blockIdx.x * blockDim.x + threadIdx.x;
    if (i < N_NODES * HID) {
        int n = i >> 6, c = i & 63;
        hs[n * (SEQ_LEN * HID) + t * HID + c] = (1.0f - Z[i]) * Ht[i];
    }
}

// LayerNorm over HID channels + ReLU. blockDim = 64, one node per block.
__global__ void k_ln_relu(const float* __restrict__ in, const float* __restrict__ gamma,
                          const float* __restrict__ beta, float* __restrict__ out) {
    __shared__ float s[HID];
    int n = blockIdx.x, c = threadIdx.x;
    float v = in[n * HID + c];
    s[c] = v; __syncthreads();
    for (int off = 32; off > 0; off >>= 1) { if (c < off) s[c] += s[c + off]; __syncthreads(); }
    float mu = s[0] * (1.0f / HID);
    __syncthreads();
    float d = v - mu;
    s[c] = d * d; __syncthreads();
    for (int off = 32; off > 0; off >>= 1) { if (c < off) s[c] += s[c + off]; __syncthreads(); }
    float var = s[0] * (1.0f / HID);
    float r = d * rsqrtf(var + 1e-5f) * gamma[c] + beta[c];
    out[n * HID + c] = fmaxf(r, 0.0f);
}

// ---------------------------------------------------------------------------
// fp32 WMMA GEMM: C[M x Nc] = act(A[M x K] @ W[K x Nc] + bias)
// One wave per 16x16 output tile, V_WMMA_F32_16X16X4_F32 over K/4 steps.
// Fragment layouts per CDNA5 ISA 7.12.2:
//   A 16x4 : lanes 0-15 -> M=lane, K=k0+{0,1}; lanes 16-31 -> M=lane-16, K=k0+{2,3}
//   B 4x16 : lanes 0-15 -> N=lane, K=k0+{0,1}; lanes 16-31 -> N=lane-16, K=k0+{2,3}
//   C/D    : VGPR r -> M = r + 8*(lane>=16), N = lane&15
// act: 0=none 1=sigmoid 2=tanh 3=relu
// ---------------------------------------------------------------------------
__global__ void k_wmma_gemm(const float* __restrict__ A, const float* __restrict__ W,
                            const float* __restrict__ bias, float* __restrict__ C,
                            int M, int K, int Nc, int act) {
    int gtid = blockIdx.x * blockDim.x + threadIdx.x;
    int wave = gtid >> 5;
    int lane = gtid & 31;
    int ntn  = Nc >> 4;
    int total = (M >> 4) * ntn;
    if (wave >= total) return;              // wave-uniform: EXEC stays all-ones
    int tm = wave / ntn, tn = wave % ntn;
    int l15   = lane & 15;
    int khalf = (lane >> 4) << 1;           // 0 or 2
    const float* arow = A + (size_t)(tm * 16 + l15) * K;
    int col = tn * 16 + l15;

    v8f acc = {0.f, 0.f, 0.f, 0.f, 0.f, 0.f, 0.f, 0.f};
    for (int k0 = 0; k0 < K; k0 += 4) {
        int ka = k0 + khalf;
        v2f a; a.x = arow[ka];            a.y = arow[ka + 1];
        v2f b; b.x = W[ka * Nc + col];    b.y = W[(ka + 1) * Nc + col];
        acc = __builtin_amdgcn_wmma_f32_16x16x4_f32(false, a, false, b,
                                                    (short)0, acc, false, false);
    }

    float bb = bias ? bias[col] : 0.0f;
    int rbase = tm * 16 + ((lane >> 4) << 3);
#pragma unroll
    for (int r = 0; r < 8; r++) {
        float v = acc[r] + bb;
        if      (act == 1) v = 1.0f / (1.0f + expf(-v));
        else if (act == 2) v = tanhf(v);
        else if (act == 3) v = fmaxf(v, 0.0f);
        C[(size_t)(rbase + r) * Nc + col] = v;
    }
}

// ---------------------------------------------------------------------------
// Interaction scores (the FLOP-dominant kernel; non-factorable due to ReLU):
//   scores[i,j] = sum_k relu(a[i,k] + b[j,k]) * w[k] + bi2   (bi1 folded into a)
// 64x64 tile per 256-thread block; each thread computes a 4x4 sub-tile.
// ---------------------------------------------------------------------------
#define TILE 64
__global__ void k_scores(const float* __restrict__ A, const float* __restrict__ B,
                         const float* __restrict__ w, const float* __restrict__ bi2,
                         float* __restrict__ out) {
    __shared__ float sA[TILE][HID + 1];   // +1: avoid 64-bank conflicts
    __shared__ float sB[TILE][HID + 1];
    __shared__ float sw[HID];
    int i0 = blockIdx.y * TILE, j0 = blockIdx.x * TILE;

    for (int idx = threadIdx.x; idx < TILE * HID; idx += 256) {
        int r = idx >> 6, c = idx & 63;
        sA[r][c] = A[(size_t)(i0 + r) * HID + c];
        sB[r][c] = B[(size_t)(j0 + r) * HID + c];
    }
    if (threadIdx.x < HID) sw[threadIdx.x] = w[threadIdx.x];
    __syncthreads();

    int jb = (threadIdx.x & 15) * 4;
    int ib = (threadIdx.x >> 4) * 4;
    float acc[4][4] = {};
#pragma unroll 4
    for (int k = 0; k < HID; k++) {
        float wk = sw[k];
        float av[4], bv[4];
#pragma unroll
        for (int r = 0; r < 4; r++) { av[r] = sA[ib + r][k]; bv[r] = sB[jb + r][k]; }
#pragma unroll
        for (int r = 0; r < 4; r++)
#pragma unroll
            for (int c = 0; c < 4; c++)
                acc[r][c] = fmaf(fmaxf(av[r] + bv[c], 0.0f), wk, acc[r][c]);
    }
    float c2 = bi2[0];
#pragma unroll
    for (int r = 0; r < 4; r++) {
        float4 o = make_float4(acc[r][0] + c2, acc[r][1] + c2,
                               acc[r][2] + c2, acc[r][3] + c2);
        *(float4*)&out[(size_t)(i0 + ib + r) * N_NODES + j0 + jb] = o;
    }
}

// ---------------------------------------------------------------------------
extern "C" void kernel_launch(void* const* d_in, const int* in_sizes, int n_in,
                              void* d_out, int out_size, void* d_ws, size_t ws_size,
                              hipStream_t stream) {
    const float* x    = (const float*)d_in[0];
    const float* ew   = (const float*)d_in[1];
    const float* Wcz  = (const float*)d_in[2];  const float* bcz = (const float*)d_in[3];
    const float* Wlz  = (const float*)d_in[4];  const float* blz = (const float*)d_in[5];
    // d_in[6..9]  Wcr/bcr/Wlr/blr: reset gate is unused (H resets to 0 each step)
    const float* Wch  = (const float*)d_in[10]; const float* bch = (const float*)d_in[11];
    const float* Wlh  = (const float*)d_in[12]; const float* blh = (const float*)d_in[13];
    const float* Wred = (const float*)d_in[14]; const float* bred= (const float*)d_in[15];
    const float* Wc1  = (const float*)d_in[16]; const float* bc1 = (const float*)d_in[17];
    const float* Wc2  = (const float*)d_in[18]; const float* bc2 = (const float*)d_in[19];
    const float* g1   = (const float*)d_in[20]; const float* be1 = (const float*)d_in[21];
    const float* g2   = (const float*)d_in[22]; const float* be2 = (const float*)d_in[23];
    const float* Wout = (const float*)d_in[24]; const float* bout= (const float*)d_in[25];
    const float* Wi1  = (const float*)d_in[26]; const float* bi1 = (const float*)d_in[27];
    const float* Wi2  = (const float*)d_in[28]; const float* bi2 = (const float*)d_in[29];
    const int*   eidx = (const int*)d_in[30];
    const int* src = eidx;
    const int* dst = eidx + E_EDGES;

    // workspace layout (floats)
    float* ws    = (float*)d_ws;
    float* deg   = ws;                    // N
    float* dinv  = deg  + N_NODES;        // N
    float* dself = dinv + N_NODES;        // N
    float* nrm   = dself+ N_NODES;        // E
    float* hw    = nrm  + E_EDGES;        // N*HID
    float* gbuf  = hw   + N_NODES * HID;  // N*HID
    float* Zb    = gbuf + N_NODES * HID;  // N*HID
    float* Hb    = Zb   + N_NODES * HID;  // N*HID
    float* hs    = Hb   + N_NODES * HID;  // N*SEQ*HID
    float* emb   = hs   + N_NODES * SEQ_LEN * HID;  // N*HID
    float* h1    = emb  + N_NODES * HID;  // N*HID
    float* h2    = h1   + N_NODES * HID;  // N*HID
    float* av    = h2   + N_NODES * HID;  // N*HID
    float* bv    = av   + N_NODES * HID;  // N*HID

    float* node_pred = (float*)d_out;                 // N x OUT
    float* scores    = node_pred + N_NODES * OUT_DIM; // N x N

    const int TPB = 256;
    const int blkNH = (N_NODES * HID) / TPB;          // 512
    const int blkEH = (E_EDGES * HID) / TPB;          // 16384

    auto gemm = [&](const float* A, const float* W, const float* b, float* C,
                    int M, int K, int Nc, int act) {
        int total  = (M >> 4) * (Nc >> 4);            // waves
        int blocks = (total * 32 + 127) / 128;
        k_wmma_gemm<<<blocks, 128, 0, stream>>>(A, W, b, C, M, K, Nc, act);
    };

    // ---- GCN normalization (shared by all convs) ----
    k_zero<<<(N_NODES + TPB - 1) / TPB, TPB, 0, stream>>>(deg, N_NODES);
    k_deg <<<E_EDGES / TPB, TPB, 0, stream>>>(dst, ew, deg);
    k_dinv<<<(N_NODES + TPB - 1) / TPB, TPB, 0, stream>>>(deg, dinv, dself);
    k_norm<<<E_EDGES / TPB, TPB, 0, stream>>>(src, dst, ew, dinv, nrm);

    // ---- TGCN timesteps (H=0 each step => h_t = (1-Z)*tanh(candidate)) ----
    for (int t = 0; t < SEQ_LEN; t++) {
        const float* xt = x + (size_t)t * N_NODES * IN_DIM;
        // update gate Z
        gemm(xt, Wcz, nullptr, hw, N_NODES, IN_DIM, HID, 0);
        k_init_g  <<<blkNH, TPB, 0, stream>>>(hw, dself, bcz, gbuf);
        k_edge_agg<<<blkEH, TPB, 0, stream>>>(src, dst, nrm, hw, gbuf);
        gemm(gbuf, Wlz, blz, Zb, N_NODES, HID, HID, 1);          // sigmoid
        // candidate H
        gemm(xt, Wch, nullptr, hw, N_NODES, IN_DIM, HID, 0);
        k_init_g  <<<blkNH, TPB, 0, stream>>>(hw, dself, bch, gbuf);
        k_edge_agg<<<blkEH, TPB, 0, stream>>>(src, dst, nrm, hw, gbuf);
        gemm(gbuf, Wlh, blh, Hb, N_NODES, HID, HID, 2);          // tanh
        k_combine<<<blkNH, TPB, 0, stream>>>(Zb, Hb, hs, t);
    }

    // ---- temporal reduction -> embeddings ----
    gemm(hs, Wred, bred, emb, N_NODES, SEQ_LEN * HID, HID, 0);

    // ---- GCN block 1 ----
    gemm(emb, Wc1, nullptr, hw, N_NODES, HID, HID, 0);
    k_init_g  <<<blkNH, TPB, 0, stream>>>(hw, dself, bc1, gbuf);
    k_edge_agg<<<blkEH, TPB, 0, stream>>>(src, dst, nrm, hw, gbuf);
    k_ln_relu <<<N_NODES, HID, 0, stream>>>(gbuf, g1, be1, h1);

    // ---- GCN block 2 ----
    gemm(h1, Wc2, nullptr, hw, N_NODES, HID, HID, 0);
    k_init_g  <<<blkNH, TPB, 0, stream>>>(hw, dself, bc2, gbuf);
    k_edge_agg<<<blkEH, TPB, 0, stream>>>(src, dst, nrm, hw, gbuf);
    k_ln_relu <<<N_NODES, HID, 0, stream>>>(gbuf, g2, be2, h2);

    // ---- node predictions ----
    gemm(h2, Wout, bout, node_pred, N_NODES, HID, OUT_DIM, 0);

    // ---- interaction predictor ----
    gemm(emb, Wi1,            bi1,     av, N_NODES, HID, HID, 0); // a + bi1 folded
    gemm(emb, Wi1 + HID * HID, nullptr, bv, N_NODES, HID, HID, 0);
    dim3 sg(N_NODES / TILE, N_NODES / TILE);
    k_scores<<<sg, 256, 0, stream>>>(av, bv, Wi2, bi2, scores);
}